// QuantumDecoder_65481071397112
// MI455X (gfx1250) — compile-verified
//
#include <hip/hip_runtime.h>

// ---------------------------------------------------------------------------
// QuantumDecoder on MI455X (gfx1250, wave32)
//
// Analytic collapse of the circuit:
//   z[b,i] = prod_{w<=i} cos(latent[b,w])        (RZ phases cancel in |amp|^2;
//                                                 CNOT chain = prefix parity)
//   out    = z @ W_map^T + b_map                 ([128,14] x [14,784])
//
// Single kernel: each wave computes one 16x16 output tile with four chained
// V_WMMA_F32_16X16X4_F32 ops (exact fp32, K padded 14 -> 16).
// All register arrays use compile-time indices only; the lane-half split is
// resolved with single v_cndmask selects (no select cascades).
// ---------------------------------------------------------------------------

typedef __attribute__((ext_vector_type(2))) float v2f;
typedef __attribute__((ext_vector_type(8))) float v8f;

#define LATENT 14
#define OUT    784
#define BATCH  128

__global__ __launch_bounds__(256) void qdec_wmma_kernel(
    const float* __restrict__ latent,  // [128, 14]
    const float* __restrict__ W_map,   // [784, 14]
    const float* __restrict__ b_map,   // [784]
    float* __restrict__ out)           // [128, 784]
{
    const int lane  = threadIdx.x & 31;
    const int wave  = threadIdx.x >> 5;      // 0..7 : M tile (16 rows each)
    const int mbase = wave * 16;
    const int nbase = blockIdx.x * 16;       // 0..48 : N tile

    const int half = lane >> 4;              // 0 = lanes 0-15, 1 = lanes 16-31
    const int l15  = lane & 15;

    // ---- B-matrix values: col N = l15 of W^T, K pair = 2*half (+4 per t) ----
    // Runtime *address* offset is fine for VMEM; only t=3/half=1 needs the
    // explicit zero pad (also avoids reading 2 floats past the end of W_map).
    const float* wrow = W_map + (nbase + l15) * LATENT;
    const int ko = 2 * half;                 // 0 or 2
    const float b00 = wrow[ko + 0], b01 = wrow[ko + 1];
    const float b10 = wrow[ko + 4], b11 = wrow[ko + 5];
    const float b20 = wrow[ko + 8], b21 = wrow[ko + 9];
    const float b30 = half ? 0.0f : wrow[12];
    const float b31 = half ? 0.0f : wrow[13];

    // ---- bias (depends only on N) ----
    const float bias = b_map[nbase + l15];

    // ---- z row for this lane's M: prefix products of cos(latent[m,*]) ----
    float zp[LATENT];
    {
        const float* lr = latent + (mbase + l15) * LATENT;
        float p = 1.0f;
#pragma unroll
        for (int k = 0; k < LATENT; ++k) {   // constant indices only
            p *= cosf(lr[k]);
            zp[k] = p;
        }
    }

    // ---- A-matrix pairs per step: one cndmask per value, no cascades ----
    const float a00 = half ? zp[2]  : zp[0],  a01 = half ? zp[3]  : zp[1];
    const float a10 = half ? zp[6]  : zp[4],  a11 = half ? zp[7]  : zp[5];
    const float a20 = half ? zp[10] : zp[8],  a21 = half ? zp[11] : zp[9];
    const float a30 = half ? 0.0f   : zp[12], a31 = half ? 0.0f   : zp[13];

    // ---- accumulator preloaded with bias ----
    v8f c;
#pragma unroll
    for (int v = 0; v < 8; ++v) c[v] = bias;

    // ---- 4 chained 16x16x4 fp32 WMMAs cover K = 0..15 ----
    v2f a, b;
    a.x = a00; a.y = a01; b.x = b00; b.y = b01;
    c = __builtin_amdgcn_wmma_f32_16x16x4_f32(false, a, false, b, (short)0, c, false, false);
    a.x = a10; a.y = a11; b.x = b10; b.y = b11;
    c = __builtin_amdgcn_wmma_f32_16x16x4_f32(false, a, false, b, (short)0, c, false, false);
    a.x = a20; a.y = a21; b.x = b20; b.y = b21;
    c = __builtin_amdgcn_wmma_f32_16x16x4_f32(false, a, false, b, (short)0, c, false, false);
    a.x = a30; a.y = a31; b.x = b30; b.y = b31;
    c = __builtin_amdgcn_wmma_f32_16x16x4_f32(false, a, false, b, (short)0, c, false, false);

    // ---- store D: VGPR v -> row M = v + 8*half, col N = l15 ----
#pragma unroll
    for (int v = 0; v < 8; ++v) {
        const int m = mbase + v + half * 8;
        out[m * OUT + nbase + l15] = c[v];
    }
}

extern "C" void kernel_launch(void* const* d_in, const int* in_sizes, int n_in,
                              void* d_out, int out_size, void* d_ws, size_t ws_size,
                              hipStream_t stream) {
    const float* latent = (const float*)d_in[0];
    // d_in[1] ("weights") is mathematically unused: diagonal RZ phases have
    // unit modulus and cancel in the probabilities.
    const float* W_map = (const float*)d_in[2];
    const float* b_map = (const float*)d_in[3];
    float* out = (float*)d_out;

    // 49 N-tiles (784/16) x 8 waves (one M-tile each, 128/16) per block.
    qdec_wmma_kernel<<<dim3(49), dim3(256), 0, stream>>>(latent, W_map, b_map, out);
}